// DimePredictor_64450279244315
// MI455X (gfx1250) — compile-verified
//
#include <hip/hip_runtime.h>

// DimeNet edge_init for MI455X (gfx1250, wave32).
// Memory-bound: ~0.9 GB HBM traffic -> ~40us floor at 23.3 TB/s.
//  - NT stores for the 672MB output stream (protect 168MB rbf_env in 192MB L2)
//  - LDS-staged, float2-vectorized, fully-coalesced phase-2 emit
//  - WMMA f32 16x16x4 evaluates the 7-term spherical basis (basis change == tiny GEMM)

#define TPB 256
#define NS 7   // NUM_SPHERICAL
#define NR 6   // NUM_RADIAL
#define SR 42  // NS*NR

typedef __attribute__((ext_vector_type(2))) float v2f;
typedef __attribute__((ext_vector_type(8))) float v8f;

// B matrix [power k][degree n]: cbf_n(c) = sum_k c^k * BTAB[k][n]
// = sqrt((2n+1)/4pi) * P_n(c) expanded in monomials. Columns n>=7 are zero.
__device__ const float BTAB[8][16] = {
  { 0.28209479f, 0.0f, -0.31539157f, 0.0f,  0.31735664f, 0.0f, -0.31784601f, 0,0,0,0,0,0,0,0,0 },
  { 0.0f, 0.48860251f, 0.0f, -1.11952900f, 0.0f,  1.75425484f, 0.0f, 0,0,0,0,0,0,0,0,0 },
  { 0.0f, 0.0f, 0.94617470f, 0.0f, -3.17356641f, 0.0f,  6.67476624f, 0,0,0,0,0,0,0,0,0 },
  { 0.0f, 0.0f, 0.0f, 1.86588166f, 0.0f, -8.18652257f, 0.0f, 0,0,0,0,0,0,0,0,0 },
  { 0.0f, 0.0f, 0.0f, 0.0f, 3.70249414f, 0.0f, -20.02429867f, 0,0,0,0,0,0,0,0,0 },
  { 0.0f, 0.0f, 0.0f, 0.0f, 0.0f, 7.36787031f, 0.0f, 0,0,0,0,0,0,0,0,0 },
  { 0.0f, 0.0f, 0.0f, 0.0f, 0.0f, 0.0f, 14.68448575f, 0,0,0,0,0,0,0,0,0 },
  { 0,0,0,0,0,0,0,0,0,0,0,0,0,0,0,0 }  // k=7 pad (multiplies zeros)
};

__global__ __launch_bounds__(TPB)
void dime_edge_init_kernel(const float* __restrict__ o,
                           const float* __restrict__ rbf,
                           const int*   __restrict__ src_idx,
                           const int*   __restrict__ dst_idx,
                           float*       __restrict__ out,
                           int T)
{
  __shared__ float s_cbf[TPB * 8];  // [triplet][degree], stride 8 (pad)
  __shared__ int   s_srcB[TPB];     // src row byte-offset into rbf (src*168)

  const int tid       = threadIdx.x;
  const int blockBase = blockIdx.x * TPB;

  // ---- Phase 1: per-triplet cos(angle), basis via WMMA ----
  int t  = blockBase + tid;
  int tc = (t < T) ? t : (T - 1);   // clamp: keep EXEC all-ones for WMMA

  const int s = __builtin_nontemporal_load(&src_idx[tc]); // streamed once: NT
  const int d = __builtin_nontemporal_load(&dst_idx[tc]);
  s_srcB[tid] = s * (SR * 4);

  const size_t sb = (size_t)s * 3, db = (size_t)d * 3;
  const float r1x = o[sb + 0], r1y = o[sb + 1], r1z = o[sb + 2];
  const float r2x = o[db + 0], r2y = o[db + 1], r2z = o[db + 2];

  const float x  = r1x * r2x + r1y * r2y + r1z * r2z;
  const float n1 = r1x * r1x + r1y * r1y + r1z * r1z;
  const float n2 = r2x * r2x + r2y * r2y + r2z * r2z;
  const float dd = n1 * n2;
  // cos(atan2(|R1 x R2|, R1.R2)) == dot/(|R1||R2|); atan2(0,0)->0 -> cos=1
  float c = (dd > 0.0f) ? x * rsqrtf(dd) : 1.0f;
  c = fminf(1.0f, fmaxf(-1.0f, c));

#if defined(__gfx1250__) && __has_builtin(__builtin_amdgcn_wmma_f32_16x16x4_f32)
  {
    const int  lane = tid & 31;
    const int  n    = lane & 15;          // output degree column
    const bool hi   = lane >= 16;
    const int  k0   = hi ? 2 : 0;
    // B operand, mirroring the ISA 16x4 A layout: v0 -> K=k0, v1 -> K=k0+1
    v2f b0; b0.x = BTAB[k0 + 0][n]; b0.y = BTAB[k0 + 1][n];
    v2f b1; b1.x = BTAB[k0 + 4][n]; b1.y = BTAB[k0 + 5][n];

    const int tbase = (tid >> 5) * 32;    // this wave's first local triplet
#pragma unroll
    for (int tile = 0; tile < 2; ++tile) {
      const float cm = __shfl(c, tile * 16 + n, 32); // broadcast triplet's cos
      const float c2 = cm * cm;
      const float c4 = c2 * c2;
      const float q  = hi ? c2 : 1.0f;
      // A layout (16x4 f32): lanes0-15 hold K=0/1, lanes16-31 hold K=2/3
      v2f a0; a0.x = q;       a0.y = q * cm;        // c^{0,1} | c^{2,3}
      v2f a1; a1.x = q * c4;  a1.y = q * c4 * cm;   // c^{4,5} | c^{6,7}

      v8f acc = {};
      acc = __builtin_amdgcn_wmma_f32_16x16x4_f32(false, a0, false, b0,
                                                  (short)0, acc, false, false);
      acc = __builtin_amdgcn_wmma_f32_16x16x4_f32(false, a1, false, b1,
                                                  (short)0, acc, false, false);
      // D layout: VGPR v = row M=v (lanes0-15) / M=v+8 (lanes16-31), N=lane&15
      if (n < NS) {
        const int rowOff = tbase + tile * 16 + (hi ? 8 : 0);
#pragma unroll
        for (int v = 0; v < 8; ++v)
          s_cbf[(rowOff + v) * 8 + n] = acc[v];
      }
    }
  }
#else
  {
    float p0 = 1.0f, p1 = c;
    s_cbf[tid * 8 + 0] = 0.28209479f;
    s_cbf[tid * 8 + 1] = 0.48860251f * c;
    const float coef[7] = { 0.28209479f, 0.48860251f, 0.63078131f,
                            0.74635267f, 0.84628438f, 0.93560258f, 1.01710724f };
#pragma unroll
    for (int l = 2; l < NS; ++l) {
      const float p = ((2 * l - 1) * c * p1 - (l - 1) * p0) / (float)l;
      s_cbf[tid * 8 + l] = coef[l] * p;
      p0 = p1; p1 = p;
    }
  }
#endif

  __syncthreads();

  // ---- Phase 2: out[t, s*6+r] = rbf[src[t], s*6+r] * cbf[t, s] ----
  const int rem = T - blockBase;
  const char* rbfB = (const char*)rbf;
  float* outTile = out + (size_t)blockBase * SR;

  if (rem >= TPB) {
    // Fast path: full 256x42 tile, float2-vectorized, unguarded.
    // Even f: pair (j, j+1) never crosses a row (j!=41) nor a degree (j%6!=5).
    const int f0 = tid * 2;
    int tl = f0 / SR;                  // one divide per thread
    int j  = f0 - tl * SR;
    v2f* outv = (v2f*)outTile + tid;   // pair index tid + 256*k
#pragma unroll
    for (int k = 0; k < (TPB * SR) / (TPB * 2); ++k) {   // 21 iterations
      const int   srcB = s_srcB[tl];
      const v2f   rv   = *(const v2f*)(rbfB + srcB + j * 4);
      const float cb   = s_cbf[tl * 8 + ((j * 43) >> 8)]; // j/6 for j<42
      v2f v; v.x = rv.x * cb; v.y = rv.y * cb;
      __builtin_nontemporal_store(v, outv + k * TPB);     // imm offsets
      // f += 512  ->  tl += 12, j += 8, wrap
      const int jn = j + 8, tn = tl + 12;
      const bool w = jn >= SR;
      j  = w ? jn - SR : jn;
      tl = w ? tn + 1  : tn;
    }
  } else {
    // Tail path (only possible final partial block): scalar, guarded.
    const int valid = rem * SR;
    for (int k = 0; k < SR; ++k) {
      const int f  = k * TPB + tid;
      const int tl = f / SR;
      const int j  = f - tl * SR;
      const float v = *(const float*)(rbfB + s_srcB[tl] + j * 4)
                      * s_cbf[tl * 8 + ((j * 43) >> 8)];
      if (f < valid)
        __builtin_nontemporal_store(v, outTile + f);
    }
  }
}

extern "C" void kernel_launch(void* const* d_in, const int* in_sizes, int n_in,
                              void* d_out, int out_size, void* d_ws, size_t ws_size,
                              hipStream_t stream) {
  (void)n_in; (void)out_size; (void)d_ws; (void)ws_size;
  const float* o   = (const float*)d_in[0];
  const float* rbf = (const float*)d_in[1];
  const int*   src = (const int*)d_in[2];
  const int*   dst = (const int*)d_in[3];
  float*       out = (float*)d_out;
  const int T = in_sizes[2];
  if (T <= 0) return;
  const int blocks = (T + TPB - 1) / TPB;
  hipLaunchKernelGGL(dime_edge_init_kernel, dim3(blocks), dim3(TPB), 0, stream,
                     o, rbf, src, dst, out, T);
}